// Loss_refine_8778913153424
// MI455X (gfx1250) — compile-verified
//
#include <hip/hip_runtime.h>
#include <math.h>

typedef float v2f __attribute__((ext_vector_type(2)));
typedef float v8f __attribute__((ext_vector_type(8)));

// ---------------------------------------------------------------------------
// Kernel 1: quat->R, pred = R*mp + t; emit
//   pred4[b,m] = {-2px,-2py,-2pz, 1.0}   (A operand; K=3 slot = 1.0)
//   t4  [b,m] = { tx,  ty,  tz,  t2 }    (B operand; K=3 slot = |t|^2)
//   => WMMA with C=0 directly yields  t2[n] - 2*pred_m . t_n
//   p2  [b,m] = |pred|^2                 (added back after the n-min)
//   add_sums[b] += sum_m ||pred-target|| (block-reduced atomic)
// ---------------------------------------------------------------------------
__global__ __launch_bounds__(256)
void prep_kernel(const float* __restrict__ pred_r,
                 const float* __restrict__ pred_t,
                 const float* __restrict__ mp,
                 const float* __restrict__ tgt,
                 float* __restrict__ pred4,
                 float* __restrict__ t4,
                 float* __restrict__ p2,
                 float* __restrict__ add_sums,
                 int M) {
    const int b = blockIdx.y;
    const int m = blockIdx.x * blockDim.x + threadIdx.x;

    // quaternion -> rotation (auto-normalized, matches reference)
    float qw = pred_r[b * 4 + 0], qx = pred_r[b * 4 + 1];
    float qy = pred_r[b * 4 + 2], qz = pred_r[b * 4 + 3];
    float inv = 1.0f / (sqrtf(qw * qw + qx * qx + qy * qy + qz * qz) + 1e-8f);
    qw *= inv; qx *= inv; qy *= inv; qz *= inv;
    const float R00 = 1.f - 2.f * (qy * qy + qz * qz);
    const float R01 = 2.f * (qx * qy - qz * qw);
    const float R02 = 2.f * (qx * qz + qy * qw);
    const float R10 = 2.f * (qx * qy + qz * qw);
    const float R11 = 1.f - 2.f * (qx * qx + qz * qz);
    const float R12 = 2.f * (qy * qz - qx * qw);
    const float R20 = 2.f * (qx * qz - qy * qw);
    const float R21 = 2.f * (qy * qz + qx * qw);
    const float R22 = 1.f - 2.f * (qx * qx + qy * qy);
    const float tx = pred_t[b * 3 + 0], ty = pred_t[b * 3 + 1], tz = pred_t[b * 3 + 2];

    const size_t idx = (size_t)b * M + m;
    const float mx = mp[idx * 3 + 0], my = mp[idx * 3 + 1], mz = mp[idx * 3 + 2];
    const float px = R00 * mx + R01 * my + R02 * mz + tx;
    const float py = R10 * mx + R11 * my + R12 * mz + ty;
    const float pz = R20 * mx + R21 * my + R22 * mz + tz;

    const float gx = tgt[idx * 3 + 0], gy = tgt[idx * 3 + 1], gz = tgt[idx * 3 + 2];

    float4 pv; pv.x = -2.f * px; pv.y = -2.f * py; pv.z = -2.f * pz; pv.w = 1.0f;
    *(float4*)(pred4 + idx * 4) = pv;
    float4 tv; tv.x = gx; tv.y = gy; tv.z = gz; tv.w = gx * gx + gy * gy + gz * gz;
    *(float4*)(t4 + idx * 4) = tv;
    p2[idx] = px * px + py * py + pz * pz;

    const float dx = px - gx, dy = py - gy, dz = pz - gz;
    float addterm = sqrtf(dx * dx + dy * dy + dz * dz);

    __shared__ float red[256];
    red[threadIdx.x] = addterm;
    __syncthreads();
    for (int s = blockDim.x >> 1; s > 0; s >>= 1) {
        if (threadIdx.x < (unsigned)s) red[threadIdx.x] += red[threadIdx.x + s];
        __syncthreads();
    }
    if (threadIdx.x == 0) atomicAdd(&add_sums[b], red[0]);
}

// ---------------------------------------------------------------------------
// Kernel 2: per 16-row m-tile, loop all n-tiles.
//   D = A([-2p | 1], 16x4) x B([t | t2]^T, 4x16) + 0
//     = t2[n] - 2*pred_m . t_n          (one V_WMMA_F32_16X16X4_F32, C inline 0)
// Inner loop = 1x global_load_b64 + 1x WMMA + componentwise running min.
// sqrt is monotone: min over d^2, add p2[m] and sqrt once per row at the end.
// ---------------------------------------------------------------------------
__global__ __launch_bounds__(128)
void adds_kernel(const float* __restrict__ pred4,
                 const float* __restrict__ t4,
                 const float* __restrict__ p2,
                 float* __restrict__ adds_sums,
                 int M) {
    const int b    = blockIdx.y;
    const int wave = threadIdx.x >> 5;
    const int lane = threadIdx.x & 31;
    const int half = lane >> 4;   // 0: K=0,1 / rows m+0..7 ; 1: K=2,3 / rows m+8..15
    const int col  = lane & 15;
    const int m_base = (blockIdx.x * 4 + wave) * 16;
    const size_t base = (size_t)b * M;

    // A-matrix (16x4 f32): lane holds row m_base+col; half selects K pair.
    const float* pa = pred4 + (base + m_base + col) * 4 + half * 2;
    v2f a; a.x = pa[0]; a.y = pa[1];

    // B-matrix (4x16 f32): lane holds column n; half 0 reads (x,y), half 1 (z,t2).
    const float* tb = t4 + (base + col) * 4 + half * 2;

    const float INF = 3.402823466e+38f;
    v8f dmin = {INF, INF, INF, INF, INF, INF, INF, INF};
    const v8f czero = {};

    const int ntiles = M >> 4;
#pragma unroll 8
    for (int nt = 0; nt < ntiles; ++nt) {
        const v2f bt = *(const v2f*)(tb + nt * 64);   // 16 points * 4 floats
        v8f d = __builtin_amdgcn_wmma_f32_16x16x4_f32(
            /*neg_a=*/false, a, /*neg_b=*/false, bt,
            /*c_mod=*/(short)0, czero, /*reuse_a=*/false, /*reuse_b=*/false);
#pragma unroll
        for (int r = 0; r < 8; ++r) dmin[r] = fminf(dmin[r], d[r]);
    }

    // Min across the 16 lanes of each half (rows r / r+8), then |pred|^2 + sqrt.
    float wsum = 0.f;
#pragma unroll
    for (int r = 0; r < 8; ++r) {
        float v = dmin[r];
        v = fminf(v, __shfl_xor(v, 1, 32));
        v = fminf(v, __shfl_xor(v, 2, 32));
        v = fminf(v, __shfl_xor(v, 4, 32));
        v = fminf(v, __shfl_xor(v, 8, 32));
        const int rowm = m_base + r + half * 8;
        const float full = p2[base + rowm] + v;
        wsum += sqrtf(fmaxf(full, 0.f));
    }
    if (col == 0) atomicAdd(&adds_sums[b], wsum);  // lanes 0 and 16
}

// ---------------------------------------------------------------------------
// Kernel 3: dists[b] = (sym ? adds : add) / M ; loss = mean(dists).
// out[0] = loss, out[1..B] = dists.
// ---------------------------------------------------------------------------
__global__ void finalize_kernel(const float* __restrict__ add_sums,
                                const float* __restrict__ adds_sums,
                                const int* __restrict__ idx,
                                float* __restrict__ out,
                                int B, int M) {
    const int b = threadIdx.x;
    float dist = 0.f;
    if (b < B) {
        const int id = idx[b];
        const bool sym = (id >= 0) && (id <= 3);   // isin(idx, [0,1,2,3])
        dist = (sym ? adds_sums[b] : add_sums[b]) * (1.0f / (float)M);
        out[1 + b] = dist;
    }
    float s = dist;
    s += __shfl_xor(s, 1, 32);
    s += __shfl_xor(s, 2, 32);
    s += __shfl_xor(s, 4, 32);
    s += __shfl_xor(s, 8, 32);
    s += __shfl_xor(s, 16, 32);
    if (b == 0) out[0] = s / (float)B;
}

extern "C" void kernel_launch(void* const* d_in, const int* in_sizes, int n_in,
                              void* d_out, int out_size, void* d_ws, size_t ws_size,
                              hipStream_t stream) {
    const float* pred_r = (const float*)d_in[0];
    const float* pred_t = (const float*)d_in[1];
    const float* mp     = (const float*)d_in[2];
    const float* tgt    = (const float*)d_in[3];
    const int*   idx    = (const int*)d_in[4];
    float* out = (float*)d_out;

    const int B = in_sizes[0] / 4;                 // 32
    const int M = in_sizes[2] / (3 * B);           // 2048

    float* ws = (float*)d_ws;
    const size_t n = (size_t)B * M;
    float* pred4     = ws;                 // n*4 floats
    float* t4        = pred4 + n * 4;      // n*4 floats
    float* p2        = t4 + n * 4;         // n floats
    float* add_sums  = p2 + n;             // B floats
    float* adds_sums = add_sums + B;       // B floats

    hipMemsetAsync(add_sums, 0, 2 * B * sizeof(float), stream);

    prep_kernel<<<dim3(M / 256, B), 256, 0, stream>>>(
        pred_r, pred_t, mp, tgt, pred4, t4, p2, add_sums, M);

    adds_kernel<<<dim3(M / 64, B), 128, 0, stream>>>(
        pred4, t4, p2, adds_sums, M);

    finalize_kernel<<<1, 32, 0, stream>>>(add_sums, adds_sums, idx, out, B, M);
}